// CRFLayer_8254927143535
// MI455X (gfx1250) — compile-verified
//
#include <hip/hip_runtime.h>
#include <stdint.h>

#define BB 128
#define TT 1024
#define KK 128
#define NTHREADS 256

typedef unsigned int v4u __attribute__((ext_vector_type(4)));
typedef int          v8i __attribute__((ext_vector_type(8)));
typedef int          v4i __attribute__((ext_vector_type(4)));

__global__ __launch_bounds__(NTHREADS, 1)
void crf_viterbi_fused(const float* __restrict__ logits,
                       const int*   __restrict__ seqlen,
                       const float* __restrict__ trans,
                       float*       __restrict__ out)
{
    // 128 KiB union: first used as trans[128][128] f32 staging (64 KiB),
    // then reused as backpointers bp[1023][128] u8 (~128 KiB).
    __shared__ float          s_union_f[32768];
    __shared__ float          s_state[KK];
    __shared__ float          s_pv[NTHREADS];
    __shared__ int            s_pi[NTHREADS];
    __shared__ unsigned char  s_tag[TT];

    const int tid = threadIdx.x;
    const int b   = blockIdx.x;
    const int j   = tid & (KK - 1);   // tag column owned
    const int h   = tid >> 7;         // which half of the i-range [h*64, h*64+64)
    const int len = seqlen[b];

    // ---- Stage transitions (128x128 f32 tile) into LDS via the Tensor Data Mover ----
#if __has_builtin(__builtin_amdgcn_tensor_load_to_lds)
    if (tid == 0) {
        uint64_t ga      = (uint64_t)(uintptr_t)trans;
        uint32_t lds_off = (uint32_t)(uintptr_t)(void*)&s_union_f[0];
        // D# group0: count=1 | lds_addr | global_addr[31:0] | global_addr[56:32], type=2
        v4u g0 = { 1u, lds_off, (uint32_t)ga,
                   (uint32_t)((ga >> 32) & 0x1FFFFFFull) | (2u << 30) };
        // D# group1: data_size=4B(code 2); tensor 128x128; tile 128x128; dim0 stride 128
        v8i g1 = { (int)(2u << 16),      // wg_mask=0, data_size=2
                   (int)(128u << 16),    // tensor_dim0 lo16 = 128
                   (int)(128u << 16),    // tensor_dim0 hi=0 | tensor_dim1 lo16 = 128
                   (int)(128u << 16),    // tensor_dim1 hi=0 | tile_dim0 = 128
                   128,                  // tile_dim1 = 128, tile_dim2 = 0
                   128,                  // tensor_dim0_stride lo32 = 128
                   (int)(16384u << 16),  // stride0 hi=0 | tensor_dim1_stride lo16 (unused)
                   0 };
        v4i g2 = { 0, 0, 0, 0 };
        v4i g3 = { 0, 0, 0, 0 };
        v8i gx = { 0, 0, 0, 0, 0, 0, 0, 0 };  // extra operand in clang-23's 6-arg form
        __builtin_amdgcn_tensor_load_to_lds(g0, g1, g2, g3, gx, 0);
    }
    __builtin_amdgcn_s_wait_tensorcnt(0);
#else
    for (int i = tid; i < KK * KK; i += NTHREADS) s_union_f[i] = trans[i];
#endif
    __syncthreads();

    // Per-thread slice of the transition matrix -> registers (column j, rows h*64..h*64+63)
    float tr[64];
    #pragma unroll
    for (int r = 0; r < 64; ++r) tr[r] = s_union_f[(h * 64 + r) * KK + j];

    const float* lgp = logits + (size_t)b * TT * KK + j;
    if (tid < KK) s_state[tid] = lgp[0];
    __syncthreads();  // state ready; everyone done reading s_union as trans

    unsigned char* s_bp = (unsigned char*)s_union_f;

    // software-pipeline the per-step logit load
    float logit_next = 0.0f;
    if (tid < KK && len > 1) logit_next = lgp[KK];

    for (int t = 1; t < len; ++t) {
        float logit_cur = logit_next;
        if (tid < KK && (t + 1) < len) logit_next = lgp[(size_t)(t + 1) * KK];

        // partial max-plus + argmax over this thread's 64 candidates (first-max wins)
        float best = -3.402823466e+38f;
        int   bidx = h * 64;
        const float4* sp = (const float4*)&s_state[h * 64];  // broadcast b128 reads
        #pragma unroll
        for (int r = 0; r < 16; ++r) {
            float4 s4 = sp[r];
            float c0 = s4.x + tr[4 * r + 0];
            float c1 = s4.y + tr[4 * r + 1];
            float c2 = s4.z + tr[4 * r + 2];
            float c3 = s4.w + tr[4 * r + 3];
            if (c0 > best) { best = c0; bidx = h * 64 + 4 * r + 0; }
            if (c1 > best) { best = c1; bidx = h * 64 + 4 * r + 1; }
            if (c2 > best) { best = c2; bidx = h * 64 + 4 * r + 2; }
            if (c3 > best) { best = c3; bidx = h * 64 + 4 * r + 3; }
        }
        s_pv[tid] = best;
        s_pi[tid] = bidx;
        __syncthreads();

        if (tid < KK) {
            float v0 = s_pv[j], v1 = s_pv[j + KK];
            int   i0 = s_pi[j], i1 = s_pi[j + KK];
            float m  = v0;
            int   mi = i0;
            if (v1 > v0) { m = v1; mi = i1; }   // strict >: lower i wins ties
            s_state[j] = m + logit_cur;          // t < len guaranteed by loop bound
            s_bp[(size_t)(t - 1) * KK + j] = (unsigned char)mi;
        }
        __syncthreads();
    }

    // ---- final argmax + backtrack: inherently serial, done LDS-resident by one lane ----
    if (tid == 0) {
        float bv = s_state[0];
        int   bi = 0;
        for (int k = 1; k < KK; ++k) {
            float v = s_state[k];
            if (v > bv) { bv = v; bi = k; }
        }
        int tag = bi;
        for (int t = TT - 1; t >= 1; --t) {
            s_tag[t] = (unsigned char)tag;
            if (t < len) tag = s_bp[(size_t)(t - 1) * KK + tag];  // bp==identity for t>=len
        }
        s_tag[0] = (unsigned char)tag;
    }
    __syncthreads();

    // ---- one-hot expansion, coalesced b128 stores (512 KiB per batch) ----
    float* outb = out + (size_t)b * TT * KK;
    for (int idx = tid * 4; idx < TT * KK; idx += NTHREADS * 4) {
        int t  = idx >> 7;
        int k0 = idx & (KK - 1);
        int tg = s_tag[t];
        float4 v;
        v.x = (k0 + 0 == tg) ? 1.0f : 0.0f;
        v.y = (k0 + 1 == tg) ? 1.0f : 0.0f;
        v.z = (k0 + 2 == tg) ? 1.0f : 0.0f;
        v.w = (k0 + 3 == tg) ? 1.0f : 0.0f;
        *(float4*)(outb + idx) = v;
    }
}

extern "C" void kernel_launch(void* const* d_in, const int* in_sizes, int n_in,
                              void* d_out, int out_size, void* d_ws, size_t ws_size,
                              hipStream_t stream) {
    const float* logits = (const float*)d_in[0];
    const int*   seqlen = (const int*)d_in[1];
    const float* trans  = (const float*)d_in[2];
    float*       out    = (float*)d_out;
    (void)in_sizes; (void)n_in; (void)out_size; (void)d_ws; (void)ws_size;

    hipLaunchKernelGGL(crf_viterbi_fused, dim3(BB), dim3(NTHREADS), 0, stream,
                       logits, seqlen, trans, out);
}